// Pointer2D_53463752901434
// MI455X (gfx1250) — compile-verified
//
#include <hip/hip_runtime.h>

// ---------------------------------------------------------------------------
// Span-pointer head (banded QA span scoring) for MI455X / gfx1250.
//
// logits[b,n] = <start[b,i],W> + <end[b,j],W> + b0 - 1e7*(1-mask_i*mask_j)
// with (i,j) the banded span indices, then softmax over n.
//
// Pass 1 (bandwidth-bound, 64 MB of fp32 embeddings -> ~2.8us floor @23.3TB/s):
//   ps[r] = dot(emb[r, 0:512],   W)
//   pe[r] = dot(emb[r, 512:1024],W)
//   done with V_WMMA_F32_16X16X4_F32 (fp32 in / fp32 acc == reference math).
// Pass 2 (tiny): logits + softmax, one workgroup per batch row.
// ---------------------------------------------------------------------------

typedef __attribute__((ext_vector_type(2))) float v2f;
typedef __attribute__((ext_vector_type(4))) float v4f;
typedef __attribute__((ext_vector_type(8))) float v8f;

#define SEQ     512
#define DHALF   512
#define DFULL   1024
#define NSPANS  4068   // 505*8 + (7+6+5+4+3+2+1)
#define NFULL   4040   // rows 0..504 contribute 8 spans each

// ---------------------------------------------------------------------------
// Kernel 1: per-position partial logits via FP32 WMMA.
// Each wave handles a tile of 16 consecutive rows (flattened b*S+s).
// A (16x4 fp32): lane L<16 -> row L, K pair {k,k+1}; lane L>=16 -> K {k+2,k+3}
// B (4x16 fp32): W chunk broadcast across all 16 columns:
//   VGPR0 = W[k + 2*(L>=16)], VGPR1 = W[k+1 + 2*(L>=16)]
// D (16x16 fp32): all columns equal; lane 0 holds rows 0..7, lane 16 rows 8..15.
// ---------------------------------------------------------------------------
__global__ __launch_bounds__(256) void gemv_wmma_f32(
    const float* __restrict__ emb,   // [B*S, 1024]
    const float* __restrict__ W,     // [512]
    float* __restrict__ ps,          // [B*S]
    float* __restrict__ pe)          // [B*S]
{
    __shared__ float lw[DHALF];
    const int tid = threadIdx.x;
    lw[tid]       = W[tid];
    lw[tid + 256] = W[tid + 256];
    __syncthreads();

    const int wave = tid >> 5;
    const int lane = tid & 31;
    const int m    = lane & 15;          // row within tile
    const int kh   = (lane >> 4) << 1;   // K sub-offset: 0 or 2

    const long tile = (long)blockIdx.x * 8 + wave;
    const long row0 = tile * 16;
    const float* __restrict__ rp = emb + (row0 + m) * DFULL;

    v8f accS = {};
    v8f accE = {};

#pragma unroll 4
    for (int k = 0; k < DHALF; k += 4) {
        const int ko = k + kh;                         // even -> 8B aligned
        v2f aS = *(const v2f*)(rp + ko);               // global_load_b64
        v2f aE = *(const v2f*)(rp + DHALF + ko);       // global_load_b64
        v2f bW = *(const v2f*)(lw + ko);               // ds_load_b64

        accS = __builtin_amdgcn_wmma_f32_16x16x4_f32(
            false, aS, false, bW, (short)0, accS, false, false);
        accE = __builtin_amdgcn_wmma_f32_16x16x4_f32(
            false, aE, false, bW, (short)0, accE, false, false);
    }

    // Column 0 of D lives in lanes 0 (M=0..7) and 16 (M=8..15).
    if (lane == 0 || lane == 16) {
        const long off = row0 + ((lane >> 4) << 3);
        v4f sLo = { accS[0], accS[1], accS[2], accS[3] };
        v4f sHi = { accS[4], accS[5], accS[6], accS[7] };
        v4f eLo = { accE[0], accE[1], accE[2], accE[3] };
        v4f eHi = { accE[4], accE[5], accE[6], accE[7] };
        *(v4f*)(ps + off)     = sLo;
        *(v4f*)(ps + off + 4) = sHi;
        *(v4f*)(pe + off)     = eLo;
        *(v4f*)(pe + off + 4) = eHi;
    }
}

// ---------------------------------------------------------------------------
// Kernel 2: banded logits + masking + softmax. One block (256 thr) per batch.
// ---------------------------------------------------------------------------
__global__ __launch_bounds__(256) void span_softmax(
    const float* __restrict__ ps, const float* __restrict__ pe,
    const int* __restrict__ tt, const int* __restrict__ am,
    const float* __restrict__ bias, float* __restrict__ out)
{
    __shared__ float l[NSPANS];
    __shared__ float sp[SEQ], se[SEQ], sm[SEQ];
    __shared__ float red[8];
    __shared__ float gmax, gsum;

    const int b   = blockIdx.x;
    const int tid = threadIdx.x;
    const float b0 = bias[0];

    for (int s = tid; s < SEQ; s += 256) {
        sp[s] = ps[b * SEQ + s];
        se[s] = pe[b * SEQ + s];
        sm[s] = (float)(tt[b * SEQ + s] * am[b * SEQ + s]);
    }
    __syncthreads();

    // ---- pass 1: compute logits, track local max --------------------------
    float lmax = -3.0e38f;
    for (int n = tid; n < NSPANS; n += 256) {
        int i, j;
        if (n < NFULL) {
            i = n >> 3;
            j = i + (n & 7);
        } else {                       // tail rows 505..511, lengths 7..1
            int t = n - NFULL;
            int len = 7;
            i = 505;
            while (t >= len) { t -= len; --len; ++i; }
            j = i + t;
        }
        float v = sp[i] + se[j] + b0 - 1.0e7f * (1.0f - sm[i] * sm[j]);
        l[n] = v;
        lmax = fmaxf(lmax, v);
    }
    // block-reduce max (wave32 shuffles, then 8 partials in LDS)
    for (int o = 16; o > 0; o >>= 1)
        lmax = fmaxf(lmax, __shfl_down(lmax, o, 32));
    if ((tid & 31) == 0) red[tid >> 5] = lmax;
    __syncthreads();
    if (tid == 0) {
        float mv = red[0];
        for (int i = 1; i < 8; ++i) mv = fmaxf(mv, red[i]);
        gmax = mv;
    }
    __syncthreads();
    const float M = gmax;

    // ---- pass 2: exp + sum ------------------------------------------------
    float lsum = 0.0f;
    for (int n = tid; n < NSPANS; n += 256) {
        float e = __expf(l[n] - M);
        l[n] = e;
        lsum += e;
    }
    for (int o = 16; o > 0; o >>= 1)
        lsum += __shfl_down(lsum, o, 32);
    if ((tid & 31) == 0) red[tid >> 5] = lsum;
    __syncthreads();
    if (tid == 0) {
        float sv = 0.0f;
        for (int i = 0; i < 8; ++i) sv += red[i];
        gsum = sv;
    }
    __syncthreads();

    // ---- pass 3: normalize + store ---------------------------------------
    const float inv = 1.0f / gsum;
    for (int n = tid; n < NSPANS; n += 256)
        out[(long)b * NSPANS + n] = l[n] * inv;
}

// ---------------------------------------------------------------------------
extern "C" void kernel_launch(void* const* d_in, const int* in_sizes, int n_in,
                              void* d_out, int out_size, void* d_ws, size_t ws_size,
                              hipStream_t stream) {
    const float* emb  = (const float*)d_in[0];  // [B,512,1024] fp32
    const int*   tt   = (const int*)  d_in[1];  // [B,512]
    const int*   am   = (const int*)  d_in[2];  // [B,512]
    const float* W    = (const float*)d_in[3];  // [512,1] fp32
    const float* bias = (const float*)d_in[4];  // [1]
    float* out = (float*)d_out;                 // [B,4068]

    const int B    = in_sizes[1] / SEQ;         // 32
    const long rows = (long)B * SEQ;            // 16384

    float* ps = (float*)d_ws;                   // [B*S]
    float* pe = ps + rows;                      // [B*S]

    // 16 rows per wave, 8 waves per block -> rows/128 blocks
    gemv_wmma_f32<<<(int)(rows / 128), 256, 0, stream>>>(emb, W, ps, pe);
    span_softmax <<<B, 256, 0, stream>>>(ps, pe, tt, am, bias, out);
}